// IBLE_12833362280950
// MI455X (gfx1250) — compile-verified
//
#include <hip/hip_runtime.h>
#include <hip/hip_bf16.h>

#define N_ENT   14541
#define N_REL   237
#define M_EDGES 272115
#define DDIM    256
#define BS      16
#define GAMMA_F 12.0f
#define BIG_F   1e8f

typedef float v2f __attribute__((ext_vector_type(2)));
typedef float v8f __attribute__((ext_vector_type(8)));

// Phase 1: dis[n,b] = sigmoid(GAMMA - sum_d |rel[b,d]*(emb[b,d]-all_emb[n,d])|)
// with dis[source_idx[b], b] forced through BIG before the sigmoid.
// One wave handles a 16-entity tile x all 16 batch columns. The d-reduction is
// done on the matrix pipe: C[16n x 16b] += A(abs-diff tile for batch b) x B(e_b one-hot cols)
// using exact-f32 V_WMMA_F32_16X16X4_F32.
__global__ __launch_bounds__(256) void dis_kernel(
    const float* __restrict__ emb, const float* __restrict__ all_emb,
    const float* __restrict__ tail_r, const int* __restrict__ source_idx,
    const int* __restrict__ relation_ids, float* __restrict__ dis_ws) {
  __shared__ float s_rel[BS * DDIM];   // rel[b,d] = tail_r[relation_ids[b], d]
  __shared__ float s_embt[BS * DDIM];  // emb[b,d] * rel[b,d]

  const int tid = threadIdx.x;
  for (int i = tid; i < BS * DDIM; i += 256) {
    const int b = i >> 8;
    const float r = tail_r[relation_ids[b] * DDIM + (i & (DDIM - 1))];
    s_rel[i] = r;
    s_embt[i] = emb[i] * r;
  }
  __syncthreads();

  const int wave = tid >> 5;
  const int lane = tid & 31;
  const int tile = blockIdx.x * 8 + wave;
  const int n0 = tile * 16;
  if (n0 >= N_ENT) return;  // wave-uniform exit: EXEC stays all-ones for WMMA

  const int row = lane & 15;          // A: M index | B/C/D: column (batch) index
  const int hi = lane >> 4;           // 0 or 1
  const int khalf = hi * 2;           // A f32 16x4 layout: lanes<16 hold K=0,1; lanes>=16 hold K=2,3
  const int nld = min(n0 + row, N_ENT - 1);  // clamp OOB rows (stores guarded later)
  const float* __restrict__ arow = all_emb + (size_t)nld * DDIM;

  v8f C = {0.f, 0.f, 0.f, 0.f, 0.f, 0.f, 0.f, 0.f};

  for (int kc = 0; kc < DDIM / 4; ++kc) {
    const int k0 = kc * 4 + khalf;
    const float a0 = arow[k0];
    const float a1 = arow[k0 + 1];
#pragma unroll
    for (int b = 0; b < BS; ++b) {
      // lanes 0-15 share one LDS address (broadcast), lanes 16-31 share another
      const float r0 = s_rel[b * DDIM + k0];
      const float r1 = s_rel[b * DDIM + k0 + 1];
      const float e0 = s_embt[b * DDIM + k0];
      const float e1 = s_embt[b * DDIM + k0 + 1];
      v2f A;
      A.x = __builtin_fabsf(e0 - r0 * a0);
      A.y = __builtin_fabsf(e1 - r1 * a1);
      const float sel = (row == b) ? 1.0f : 0.0f;  // one-hot column b, all K rows
      v2f Bm;
      Bm.x = sel;
      Bm.y = sel;
      C = __builtin_amdgcn_wmma_f32_16x16x4_f32(false, A, false, Bm,
                                                (short)0, C, false, false);
    }
  }

  // C/D layout: VGPR r -> M = r (lanes 0-15) or r+8 (lanes 16-31); column = lane&15
  const int bcol = row;
  const int src = source_idx[bcol];
#pragma unroll
  for (int r = 0; r < 8; ++r) {
    const int m = n0 + r + hi * 8;
    float v = C[r];
    if (m == src) v = BIG_F;                 // self-mask
    v = 1.0f / (1.0f + __expf(v - GAMMA_F)); // sigmoid(GAMMA - dis)
    if (m < N_ENT) dis_ws[m * BS + bcol] = v;
  }
}

__global__ void zero_kernel(float* __restrict__ out, int n) {
  const int i = blockIdx.x * blockDim.x + threadIdx.x;
  if (i < n) out[i] = 0.0f;
}

// Phase 2: 16 lanes per edge (lane = batch column b).
// out[b, tails[m]] += dis[heads[m], b] * (edge_rels[m] == relation_ids[b])
__global__ __launch_bounds__(256) void edge_kernel(
    const int* __restrict__ heads, const int* __restrict__ tails,
    const int* __restrict__ edge_rels, const int* __restrict__ relation_ids,
    const float* __restrict__ dis_ws, float* __restrict__ out) {
  __shared__ int s_rid[BS];
  if (threadIdx.x < BS) s_rid[threadIdx.x] = relation_ids[threadIdx.x];
  __syncthreads();

  const long long t = (long long)blockIdx.x * blockDim.x + threadIdx.x;
  const int m = (int)(t >> 4);
  const int b = (int)(t & 15);
  if (m >= M_EDGES) return;
  if (edge_rels[m] != s_rid[b]) return;
  const float v = dis_ws[(size_t)heads[m] * BS + b];  // 64B-coalesced row gather
  atomicAdd(&out[(size_t)b * N_ENT + tails[m]], v);
}

extern "C" void kernel_launch(void* const* d_in, const int* in_sizes, int n_in,
                              void* d_out, int out_size, void* d_ws, size_t ws_size,
                              hipStream_t stream) {
  (void)in_sizes; (void)n_in; (void)out_size; (void)ws_size;
  const float* emb          = (const float*)d_in[0];
  const float* all_emb      = (const float*)d_in[1];
  const float* tail_r       = (const float*)d_in[2];
  const int*   source_idx   = (const int*)d_in[3];
  const int*   relation_ids = (const int*)d_in[4];
  const int*   heads        = (const int*)d_in[5];
  const int*   tails        = (const int*)d_in[6];
  const int*   edge_rels    = (const int*)d_in[7];
  float* out    = (float*)d_out;
  float* dis_ws = (float*)d_ws;  // N_ENT * BS floats (~931 KB)

  const int tiles = (N_ENT + 15) / 16;   // 909 entity tiles
  const int blocks1 = (tiles + 7) / 8;   // 8 waves (tiles) per 256-thread block
  dis_kernel<<<blocks1, 256, 0, stream>>>(emb, all_emb, tail_r, source_idx,
                                          relation_ids, dis_ws);

  const int n_out = BS * N_ENT;
  zero_kernel<<<(n_out + 255) / 256, 256, 0, stream>>>(out, n_out);

  const long long total = (long long)M_EDGES * BS;
  const int blocks2 = (int)((total + 255) / 256);
  edge_kernel<<<blocks2, 256, 0, stream>>>(heads, tails, edge_rels, relation_ids,
                                           dis_ws, out);
}